// ComposeRt_9783935500895
// MI455X (gfx1250) — compile-verified
//
#include <hip/hip_runtime.h>
#include <stdint.h>

// ComposeRt: inclusive scan of affine transforms along N=32.
// out[b,0] = x[b,0]; out[b,n] = compose(out[b,n-1], x[b,n])
// compose(A,B): R = Ra*Rb ; t = ta + Ra*tb   (homogeneous prefix product)
//
// Memory-bound: 402 MB traffic, ~17us floor @ 23.3 TB/s (AI ~0.76 flop/byte).
// CDNA5 path: async global->LDS DMA (ASYNCcnt), software-pipelined 2-deep so
// the DMA of chunk c+1 overlaps compose+store of chunk c. Async loads use the
// GVS addressing mode (uniform SGPR base + 32-bit per-lane VGPR offset) to
// halve address-register pressure. Output via non-temporal b128 stores
// (STOREcnt) so ASYNCcnt stays loads-only and s_wait_asynccnt 12 precisely
// bounds the in-flight prefetch (async loads retire in order).

#define THREADS      128
#define SEQ_N        32
#define CH           4                        // scan steps per chunk
#define NCH          (SEQ_N / CH)             // 8 chunks
#define F4_PER_STEP  3                        // 3x4 floats = 3 float4 / transform
#define F4_PER_CH    (CH * F4_PER_STEP)       // 12 float4 per seq per chunk
#define SLOT_F4      (F4_PER_CH + 1)          // 13: 16B pad -> conflict-light LDS
#define F4_PER_SEQ   (SEQ_N * F4_PER_STEP)    // 96 float4 per sequence in global
#define BUF_F4       (THREADS * SLOT_F4)      // 1664 float4 per buffer
#define BUF_BYTES    (BUF_F4 * 16)            // 26,624 B per buffer (x2 = 53 KB)
#define CH_BYTES     (F4_PER_CH * 16)         // 192 B per sequence per chunk

typedef __attribute__((ext_vector_type(4))) float v4f;   // native vector for NT store

__device__ __forceinline__ float4 f4fma(float s, float4 v, float4 a) {
    a.x = fmaf(s, v.x, a.x);
    a.y = fmaf(s, v.y, a.y);
    a.z = fmaf(s, v.z, a.z);
    a.w = fmaf(s, v.w, a.w);
    return a;
}

// One output row of compose(C, X): row_i = c.x*X0 + c.y*X1 + c.z*X2 + (0,0,0,c.w)
// (X row k = (r_k0, r_k1, r_k2, t_k), so the .w lane carries t correctly.)
__device__ __forceinline__ float4 composeRow(float4 c, float4 x0, float4 x1, float4 x2) {
    float4 acc = make_float4(0.0f, 0.0f, 0.0f, c.w);
    acc = f4fma(c.x, x0, acc);
    acc = f4fma(c.y, x1, acc);
    acc = f4fma(c.z, x2, acc);
    return acc;
}

__global__ void __launch_bounds__(THREADS)
ComposeRt_9783935500895_kernel(const float* __restrict__ xin,
                               float* __restrict__ xout) {
    __shared__ float4 smem[2 * BUF_F4];        // double buffer, 53,248 B

    const int tid = threadIdx.x;
    const long long b0 = (long long)blockIdx.x * THREADS;

    // Generic LDS pointer: low 32 bits are the workgroup-relative LDS byte
    // address -- what the async DMA ops take.
    const unsigned lds_base = (unsigned)(uintptr_t)(&smem[0]);

    // Per-thread cooperative tile mapping (hoisted once): linear float4 index
    // g over the block-chunk tile; consecutive lanes hit consecutive 16B.
    // All global addressing = uniform base (SGPR) + 32-bit offset (< 2^31).
    int      lof[F4_PER_CH];                   // LDS float4 offset within buffer
    unsigned gof[F4_PER_CH];                   // 32-bit global byte offset, chunk 0
#pragma unroll
    for (int i = 0; i < F4_PER_CH; ++i) {
        const int g   = i * THREADS + tid;
        const int seq = g / F4_PER_CH;
        const int q   = g - seq * F4_PER_CH;
        lof[i] = seq * SLOT_F4 + q;
        gof[i] = (unsigned)(((b0 + seq) * (long long)F4_PER_SEQ + q) * 16ll);
    }

    // ---- prologue: async DMA chunk 0 -> buffer 0 (GVS mode, NT stream) ----
#pragma unroll
    for (int i = 0; i < F4_PER_CH; ++i) {
        const unsigned la = lds_base + (unsigned)lof[i] * 16u;
        asm volatile("global_load_async_to_lds_b128 %0, %1, %2 th:TH_LOAD_NT"
                     :: "v"(la), "v"(gof[i]), "s"(xin) : "memory");
    }

    float4 c0, c1, c2;          // running prefix (carry), 3 rows of [R|t]
    bool have = false;

    for (int ch = 0; ch < NCH; ++ch) {
        const int cur = ch & 1;

        // ---- issue async DMA for chunk ch+1 into the other buffer, then
        //      wait until only those 12 remain (loads retire in order) ----
        if (ch + 1 < NCH) {
            const unsigned bsel = (unsigned)((cur ^ 1) * BUF_BYTES);
            const unsigned goff = (unsigned)((ch + 1) * CH_BYTES);
#pragma unroll
            for (int i = 0; i < F4_PER_CH; ++i) {
                const unsigned la = lds_base + bsel + (unsigned)lof[i] * 16u;
                asm volatile("global_load_async_to_lds_b128 %0, %1, %2 th:TH_LOAD_NT"
                             :: "v"(la), "v"(gof[i] + goff), "s"(xin) : "memory");
            }
            asm volatile("s_wait_asynccnt 12" ::: "memory");
        } else {
            asm volatile("s_wait_asynccnt 0" ::: "memory");
        }
        __syncthreads();        // chunk ch fully resident in buffer `cur`

        // ---- per-thread serial scan on its own slot (carry in registers) ----
        float4* slot = &smem[cur * BUF_F4 + tid * SLOT_F4];
#pragma unroll
        for (int n = 0; n < CH; ++n) {
            const float4 x0 = slot[n * 3 + 0];
            const float4 x1 = slot[n * 3 + 1];
            const float4 x2 = slot[n * 3 + 2];
            if (!have) {
                c0 = x0; c1 = x1; c2 = x2;
                have = true;
            } else {
                const float4 r0 = composeRow(c0, x0, x1, x2);
                const float4 r1 = composeRow(c1, x0, x1, x2);
                const float4 r2 = composeRow(c2, x0, x1, x2);
                c0 = r0; c1 = r1; c2 = r2;
            }
            slot[n * 3 + 0] = c0;
            slot[n * 3 + 1] = c1;
            slot[n * 3 + 2] = c2;
        }
        __syncthreads();        // results for chunk ch visible block-wide

        // ---- cooperative coalesced non-temporal store (STOREcnt path) ----
        const unsigned soff = (unsigned)(ch * CH_BYTES);
#pragma unroll
        for (int i = 0; i < F4_PER_CH; ++i) {
            const v4f v = *(const v4f*)(&smem[cur * BUF_F4 + lof[i]]);
            v4f* op = (v4f*)((char*)xout + (size_t)(gof[i] + soff));
            __builtin_nontemporal_store(v, op);
        }
        // All LDS reads of buffer `cur` are complete once every wave passes
        // this barrier (the dependent global stores force the DScnt wait),
        // so next iteration may safely DMA chunk ch+2 into buffer `cur`.
        __syncthreads();
    }
}

extern "C" void kernel_launch(void* const* d_in, const int* in_sizes, int n_in,
                              void* d_out, int out_size, void* d_ws, size_t ws_size,
                              hipStream_t stream) {
    (void)n_in; (void)out_size; (void)d_ws; (void)ws_size;
    const float* x = (const float*)d_in[0];
    float* out = (float*)d_out;

    const int nseq = in_sizes[0] / (SEQ_N * 12);       // 131072 sequences
    const int blocks = nseq / THREADS;                 // 1024 blocks

    ComposeRt_9783935500895_kernel<<<blocks, THREADS, 0, stream>>>(x, out);
}